// MVMap_43465069035945
// MI455X (gfx1250) — compile-verified
//
#include <hip/hip_runtime.h>
#include <math.h>

#define NCAMS 6
#define CCH   32
#define XVDIM 56
#define ZVDIM 100
#define IMH   224
#define IMW   400
#define YD    8
#define XD    256
#define ZD    256
#define NVOX  (YD * XD * ZD)          /* 524288 */
#define FEAT_PLANE (XVDIM * ZVDIM)    /* 5600 */
#define IMG_PLANE  (IMH * IMW)        /* 89600 */

typedef __attribute__((ext_vector_type(2))) float v2f;
typedef __attribute__((ext_vector_type(8))) float v8f;

__device__ __forceinline__ int iminc(int a, int b) { return a < b ? a : b; }
__device__ __forceinline__ int imaxc(int a, int b) { return a > b ? a : b; }

// Bilinear tap offsets/weights, reference semantics: clamp index, fold
// per-tap validity into the weight. "x" is the strided dim (Wf), "y" is unit stride (Hf).
__device__ __forceinline__ void bilin_taps(float fx, float fy, int Wd, int Hd, int strideX,
                                           int o[4], float w[4]) {
    float x0f = floorf(fx), y0f = floorf(fy);
    float wx = fx - x0f, wy = fy - y0f;
    int ix0 = (int)x0f, iy0 = (int)y0f;
    int ix1 = ix0 + 1, iy1 = iy0 + 1;
    float vx0 = (ix0 >= 0 && ix0 < Wd) ? 1.f : 0.f;
    float vx1 = (ix1 >= 0 && ix1 < Wd) ? 1.f : 0.f;
    float vy0 = (iy0 >= 0 && iy0 < Hd) ? 1.f : 0.f;
    float vy1 = (iy1 >= 0 && iy1 < Hd) ? 1.f : 0.f;
    int cx0 = iminc(imaxc(ix0, 0), Wd - 1), cx1 = iminc(imaxc(ix1, 0), Wd - 1);
    int cy0 = iminc(imaxc(iy0, 0), Hd - 1), cy1 = iminc(imaxc(iy1, 0), Hd - 1);
    o[0] = cx0 * strideX + cy0; w[0] = (1.f - wx) * (1.f - wy) * vx0 * vy0;
    o[1] = cx1 * strideX + cy0; w[1] = wx * (1.f - wy) * vx1 * vy0;
    o[2] = cx0 * strideX + cy1; w[2] = (1.f - wx) * wy * vx0 * vy1;
    o[3] = cx1 * strideX + cy1; w[3] = wx * wy * vx1 * vy1;
}

__global__ __launch_bounds__(256) void mvmap_lift_kernel(
    const float* __restrict__ imgf,     // [6,32,56,100]
    const float* __restrict__ varr,     // [6,1,224,400]
    const float* __restrict__ rdep,     // [6,1,224,400]
    const float* __restrict__ scoords,  // [1,N,3]
    const float* __restrict__ Ps,       // [6,4,4]
    const float* __restrict__ Ks,       // [6,3,3]
    float* __restrict__ out)
{
    __shared__ float Ms[NCAMS][3][4];   // combined K @ inv(P)[:3,:]
    __shared__ float uvw_s[256][18];    // per-voxel projected uvw for 6 cams

    const int tid = threadIdx.x;

    // ---- per-camera combined projection matrix (rigid-pose inverse) ----
    if (tid < NCAMS) {
        const float* P = Ps + tid * 16;
        const float* K = Ks + tid * 9;
        float R[3][3], t[3];
        #pragma unroll
        for (int i = 0; i < 3; i++) {
            #pragma unroll
            for (int j = 0; j < 3; j++) R[i][j] = P[i * 4 + j];
            t[i] = P[i * 4 + 3];
        }
        float wt[3];
        #pragma unroll
        for (int k = 0; k < 3; k++)
            wt[k] = -(R[0][k] * t[0] + R[1][k] * t[1] + R[2][k] * t[2]);
        #pragma unroll
        for (int i = 0; i < 3; i++) {
            #pragma unroll
            for (int j = 0; j < 3; j++)   // w2c_R[k][j] = R[j][k]
                Ms[tid][i][j] = K[i * 3 + 0] * R[j][0] + K[i * 3 + 1] * R[j][1] + K[i * 3 + 2] * R[j][2];
            Ms[tid][i][3] = K[i * 3 + 0] * wt[0] + K[i * 3 + 1] * wt[1] + K[i * 3 + 2] * wt[2];
        }
    }
    __syncthreads();

    const int lane = tid & 31;
    const int wave = tid >> 5;
    const int blockBase = blockIdx.x * 256;
    const bool lo = lane < 16;
    const int nB = lane & 15;
    const int k0 = lo ? 0 : 2;

    // ---- B matrices for V_WMMA_F32_16X16X4_F32 ----
    // B[k][n] = M[cam=n/3][comp=n%3][k]; cols 0..15 in wmma#1, cols 16..17 in wmma#2.
    v2f b1, b2;
    {
        int cam = nB / 3, comp = nB - cam * 3;
        b1.x = Ms[cam][comp][k0];
        b1.y = Ms[cam][comp][k0 + 1];
        int col2 = 16 + nB;
        if (col2 < 18) { int c2 = col2 % 3; b2.x = Ms[5][c2][k0]; b2.y = Ms[5][c2][k0 + 1]; }
        else           { b2.x = 0.f; b2.y = 0.f; }
    }

    // ---- batched projection via WMMA: D[p][n] = sum_k pts_h[p][k] * B[k][n] ----
    const v8f cz = {};
    #pragma unroll
    for (int g = 0; g < 2; g++) {
        int m = blockBase + wave * 32 + g * 16 + nB;     // point this lane supplies
        const float* sc = scoords + (size_t)m * 3;
        v2f a;                                            // A: lanes<16 -> K0,K1 ; lanes>=16 -> K2,K3
        a.x = lo ? sc[0] : sc[2];
        a.y = lo ? sc[1] : 1.0f;
        v8f d1 = __builtin_amdgcn_wmma_f32_16x16x4_f32(false, a, false, b1, (short)0, cz, false, false);
        v8f d2 = __builtin_amdgcn_wmma_f32_16x16x4_f32(false, a, false, b2, (short)0, cz, false, false);
        int mrow = wave * 32 + g * 16 + (lo ? 0 : 8);     // D layout: vgpr r -> row r (+8 for hi lanes)
        #pragma unroll
        for (int r = 0; r < 8; r++) uvw_s[mrow + r][nB] = d1[r];
        if (nB < 2) {
            #pragma unroll
            for (int r = 0; r < 8; r++) uvw_s[mrow + r][16 + nB] = d2[r];
        }
    }
    __syncthreads();

    // ---- per-voxel gather / masked mean ----
    const int m = blockBase + tid;
    const int z = m & 255;
    const int x = (m >> 8) & 255;
    const int y = m >> 16;

    float acc[CCH];
    #pragma unroll
    for (int c = 0; c < CCH; c++) acc[c] = 0.f;
    float accV = 0.f, accD = 0.f, accN = 0.f, cnt = 0.f;

    #pragma unroll 1
    for (int cam = 0; cam < NCAMS; cam++) {
        float ur = uvw_s[tid][cam * 3 + 0];
        float vr = uvw_s[tid][cam * 3 + 1];
        float zc = uvw_s[tid][cam * 3 + 2];
        float zsafe = (fabsf(zc) < 1e-6f) ? 1e-6f : zc;
        float u = ur / zsafe;
        float v = vr / zsafe;
        if (u != u) u = -1.f;   // nan_to_num(nan=-1)
        if (v != v) v = -1.f;
        bool mask = (zc > 1e-3f) && (u >= 0.f) && (u < (float)IMW) && (v >= 0.f) && (v < (float)IMH);
        if (!mask) continue;
        cnt  += 1.f;
        accD += zc;

        float gx = v * (2.0f / (float)IMH) - 1.0f;
        float gy = u * (2.0f / (float)IMW) - 1.0f;

        // feature sample: permuted feat [ZV,XV] -> Wf=56 (stride 100), Hf=100 (stride 1)
        {
            float fx = (gx + 1.f) * (0.5f * XVDIM) - 0.5f;
            float fy = (gy + 1.f) * (0.5f * ZVDIM) - 0.5f;
            int o[4]; float w[4];
            bilin_taps(fx, fy, XVDIM, ZVDIM, ZVDIM, o, w);
            const float* base = imgf + (size_t)cam * (CCH * FEAT_PLANE);
            #pragma unroll
            for (int c = 0; c < CCH; c++) {
                const float* p = base + (size_t)c * FEAT_PLANE;
                acc[c] += p[o[0]] * w[0] + p[o[1]] * w[1] + p[o[2]] * w[2] + p[o[3]] * w[3];
            }
        }
        // variance / depth sample: permuted [W,H] -> Wf=224 (stride 400), Hf=400 (stride 1)
        {
            float fx = (gx + 1.f) * (0.5f * IMH) - 0.5f;
            float fy = (gy + 1.f) * (0.5f * IMW) - 0.5f;
            int o[4]; float w[4];
            bilin_taps(fx, fy, IMH, IMW, IMW, o, w);
            const float* vb = varr + (size_t)cam * IMG_PLANE;
            const float* db = rdep + (size_t)cam * IMG_PLANE;
            accV += vb[o[0]] * w[0] + vb[o[1]] * w[1] + vb[o[2]] * w[2] + vb[o[3]] * w[3];
            accN += db[o[0]] * w[0] + db[o[1]] * w[1] + db[o[2]] * w[2] + db[o[3]] * w[3];
        }
    }

    const float inv = 1.0f / (cnt + 1e-6f);
    const size_t vox = (size_t)m;
    // out_feats [C*Y, X, Z]
    #pragma unroll
    for (int c = 0; c < CCH; c++) {
        size_t off = ((size_t)(c * YD + y)) * (XD * ZD) + ((size_t)x << 8) + (size_t)z;
        __builtin_nontemporal_store(acc[c] * inv, out + off);
    }
    const size_t base1 = (size_t)NVOX * CCH;
    __builtin_nontemporal_store(accV * inv, out + base1 + vox);                    // var_info
    __builtin_nontemporal_store(accD * inv, out + base1 + (size_t)NVOX + vox);     // depth
    __builtin_nontemporal_store(accN * inv, out + base1 + 2 * (size_t)NVOX + vox); // nerf_depth
}

extern "C" void kernel_launch(void* const* d_in, const int* in_sizes, int n_in,
                              void* d_out, int out_size, void* d_ws, size_t ws_size,
                              hipStream_t stream) {
    (void)in_sizes; (void)n_in; (void)out_size; (void)d_ws; (void)ws_size;
    const float* imgf = (const float*)d_in[0];
    const float* varr = (const float*)d_in[1];
    const float* rdep = (const float*)d_in[2];
    const float* sc   = (const float*)d_in[3];
    const float* Ps   = (const float*)d_in[4];
    const float* Ks   = (const float*)d_in[5];
    float* out = (float*)d_out;
    dim3 grid(NVOX / 256), block(256);
    hipLaunchKernelGGL(mvmap_lift_kernel, grid, block, 0, stream,
                       imgf, varr, rdep, sc, Ps, Ks, out);
}